// GAAEncoder_43516608643615
// MI455X (gfx1250) — compile-verified
//
#include <hip/hip_runtime.h>
#include <hip/hip_bf16.h>
#include <math.h>

#define SLOPE 0.2f

typedef float v2f __attribute__((ext_vector_type(2)));
typedef float v8f __attribute__((ext_vector_type(8)));

// ---------------------------------------------------------------- utilities

__global__ void fill_f32(float* __restrict__ p, float v, int n) {
    int i = blockIdx.x * blockDim.x + threadIdx.x;
    if (i < n) p[i] = v;
}

// Float atomic max via sign-split int/uint atomics (bit order matches float
// order within each sign class; init must be -inf = 0xFF800000).
__device__ __forceinline__ void atomicMaxFloat(float* addr, float value) {
    if (value >= 0.0f) {
        atomicMax((int*)addr, __float_as_int(value));
    } else {
        atomicMin((unsigned int*)addr, __float_as_uint(value));
    }
}

__device__ __forceinline__ float leaky(float v) {
    return v > 0.0f ? v : SLOPE * v;
}

// --------------------------------------------------- fp32 WMMA strip GEMM
// C[M,N] = A[M,K] * B[K,N], all row-major fp32, N = NT*16, K % 4 == 0.
// B (the small weight matrix) is staged into LDS once per block; each wave
// owns a 16-row strip across ALL N columns so every A fetch (one b64 per
// lane per k-step) feeds NT back-to-back V_WMMA_F32_16X16X4_F32 ops.
//   A 16x4 layout : lanes 0-15 hold K={0,1}, lanes 16-31 hold K={2,3}; M = lane%16
//   B 4x16 layout : mirror (N = lane%16)
//   C/D 16x16     : VGPR v -> row v (lanes 0-15) / row v+8 (lanes 16-31), col = lane%16
template <int NT>
__global__ void gemm_wmma_strip(const float* __restrict__ A,
                                const float* __restrict__ B,
                                float* __restrict__ C,
                                int M, int K) {
    const int N = NT * 16;
    extern __shared__ float sB[];             // K * N floats

    // cooperative stage of B into LDS (K*N is a multiple of blockDim here)
    for (int i = threadIdx.x; i < K * N; i += blockDim.x) sB[i] = B[i];
    __syncthreads();

    const int lane = threadIdx.x & 31;
    const int wid  = threadIdx.x >> 5;
    const int row0 = (blockIdx.x * (blockDim.x >> 5) + wid) * 16;
    if (row0 >= M) return;                    // wave-uniform exit (EXEC all-1s)

    const int mn = lane & 15;
    const int hi = lane >> 4;
    const int kg = hi << 1;                   // K pair this lane half covers

    const float* arow = A + (size_t)(row0 + mn) * K;

    v8f acc[NT];
#pragma unroll
    for (int t = 0; t < NT; ++t) acc[t] = (v8f){};

    for (int k0 = 0; k0 < K; k0 += 4) {
        v2f a;
        a.x = arow[k0 + kg];
        a.y = arow[k0 + kg + 1];
        const float* brow = sB + (size_t)(k0 + kg) * N + mn;
#pragma unroll
        for (int t = 0; t < NT; ++t) {
            v2f b;
            b.x = brow[t * 16];
            b.y = brow[t * 16 + N];
            acc[t] = __builtin_amdgcn_wmma_f32_16x16x4_f32(
                /*neg_a=*/false, a, /*neg_b=*/false, b,
                /*c_mod=*/(short)0, acc[t], /*reuse_a=*/false, /*reuse_b=*/false);
        }
    }

#pragma unroll
    for (int t = 0; t < NT; ++t) {
        float* crow = C + (size_t)(row0 + hi * 8) * N + t * 16 + mn;
#pragma unroll
        for (int v = 0; v < 8; ++v) crow[(size_t)v * N] = acc[t][v];
    }
}

// ------------------------------------------- per-node attention scores GEMV
// One wave per node: ls[n] = h[n,:].a_src, ld[n] = h[n,:].a_dst
__global__ void node_scores(const float* __restrict__ h,
                            const float* __restrict__ a_src,
                            const float* __restrict__ a_dst,
                            float* __restrict__ ls, float* __restrict__ ld,
                            int n_nodes, int C) {
    const int lane = threadIdx.x & 31;
    const int node = (blockIdx.x * blockDim.x + threadIdx.x) >> 5;
    if (node >= n_nodes) return;
    float s = 0.0f, d = 0.0f;
    for (int c = lane; c < C; c += 32) {
        float v = h[(size_t)node * C + c];
        s += v * a_src[c];
        d += v * a_dst[c];
    }
#pragma unroll
    for (int off = 16; off > 0; off >>= 1) {
        s += __shfl_xor(s, off, 32);
        d += __shfl_xor(d, off, 32);
    }
    if (lane == 0) { ls[node] = s; ld[node] = d; }
}

// -------------------------------------------------------------- edge passes

// Pass 1: compute leaky-relu logit, stash it in ev[e], segment max by dst.
__global__ void edge_logit_max(const int* __restrict__ src, const int* __restrict__ dst,
                               const float* __restrict__ ls, const float* __restrict__ ld,
                               float* __restrict__ ev, float* __restrict__ lmax, int E) {
    int e = blockIdx.x * blockDim.x + threadIdx.x;
    if (e >= E) return;
    int d = dst[e];
    float lg = leaky(ls[src[e]] + ld[d]);
    ev[e] = lg;
    atomicMaxFloat(&lmax[d], lg);
}

// Pass 2: ev[e] <- exp(logit - lmax[dst]); segment sum into denom.
__global__ void edge_expsum(const int* __restrict__ dst,
                            const float* __restrict__ lmax,
                            float* __restrict__ ev, float* __restrict__ denom, int E) {
    int e = blockIdx.x * blockDim.x + threadIdx.x;
    if (e >= E) return;
    int d = dst[e];
    float ex = __expf(ev[e] - lmax[d]);
    ev[e] = ex;
    atomicAdd(&denom[d], ex);
}

// Pass 3: C/4 threads per edge; each thread moves a float4 of h[src] scaled
// by alpha into agg[dst] via 4 float atomics (L2-resident on 192MB L2).
template <int C>
__global__ void edge_aggregate(const int* __restrict__ src, const int* __restrict__ dst,
                               const float* __restrict__ ev, const float* __restrict__ denom,
                               const float* __restrict__ h, float* __restrict__ agg, int E) {
    constexpr int GP = C / 4;
    int t = blockIdx.x * blockDim.x + threadIdx.x;
    int e = t / GP;
    int g = t - e * GP;
    if (e >= E) return;
    int s = src[e], d = dst[e];
    float alpha = ev[e] / denom[d];
    const float4 v = *((const float4*)(h + (size_t)s * C) + g);
    float* ap = agg + (size_t)d * C + g * 4;
    atomicAdd(ap + 0, alpha * v.x);
    atomicAdd(ap + 1, alpha * v.y);
    atomicAdd(ap + 2, alpha * v.z);
    atomicAdd(ap + 3, alpha * v.w);
}

// ------------------------------------------------------------- epilogues

__global__ void bias_act(const float* __restrict__ agg, const float* __restrict__ b,
                         float* __restrict__ out, int total, int C, int do_relu) {
    int i = blockIdx.x * blockDim.x + threadIdx.x;
    if (i >= total) return;
    float v = agg[i] + b[i % C];
    out[i] = do_relu ? fmaxf(v, 0.0f) : v;
}

// ---------------------------------------------------------------- driver

static inline int cdiv(int a, int b) { return (a + b - 1) / b; }

extern "C" void kernel_launch(void* const* d_in, const int* in_sizes, int n_in,
                              void* d_out, int out_size, void* d_ws, size_t ws_size,
                              hipStream_t stream) {
    const float* x      = (const float*)d_in[0];
    const int*   eidx   = (const int*)d_in[1];
    const float* W1     = (const float*)d_in[2];
    const float* a_src1 = (const float*)d_in[3];
    const float* a_dst1 = (const float*)d_in[4];
    const float* b1     = (const float*)d_in[5];
    const float* W2     = (const float*)d_in[6];
    const float* a_src2 = (const float*)d_in[7];
    const float* a_dst2 = (const float*)d_in[8];
    const float* b2     = (const float*)d_in[9];
    float* out = (float*)d_out;

    const int IN = 128, H = 64, O = 32;
    const int N = in_sizes[0] / IN;          // 50000
    const int E = in_sizes[1] / 2;           // 800000
    const int* src = eidx;
    const int* dst = eidx + E;

    // workspace layout (floats)
    float* ws    = (float*)d_ws;
    float* h1    = ws;                        // N*H  (layer-1 projection, then reused as layer-2 input)
    float* agg1  = h1 + (size_t)N * H;        // N*H
    float* h2    = agg1 + (size_t)N * H;      // N*O
    float* ls    = h2 + (size_t)N * O;        // N
    float* ld    = ls + N;                    // N
    float* lmax  = ld + N;                    // N
    float* denom = lmax + N;                  // N
    float* ev    = denom + N;                 // E

    const int B = 256;                        // 8 waves -> 128 rows per GEMM block

    // ---------------- layer 1 ----------------
    // projection: h1 = x @ W1   (fp32 WMMA, W1 staged in 32KB LDS)
    gemm_wmma_strip<4><<<cdiv(N, 128), B, IN * H * sizeof(float), stream>>>(x, W1, h1, N, IN);

    node_scores<<<cdiv(N * 32, B), B, 0, stream>>>(h1, a_src1, a_dst1, ls, ld, N, H);

    fill_f32<<<cdiv(N, B), B, 0, stream>>>(lmax, -INFINITY, N);
    fill_f32<<<cdiv(N, B), B, 0, stream>>>(denom, 0.0f, N);
    fill_f32<<<cdiv(N * H, B), B, 0, stream>>>(agg1, 0.0f, N * H);

    edge_logit_max<<<cdiv(E, B), B, 0, stream>>>(src, dst, ls, ld, ev, lmax, E);
    edge_expsum<<<cdiv(E, B), B, 0, stream>>>(dst, lmax, ev, denom, E);
    edge_aggregate<64><<<cdiv(E * (H / 4), B), B, 0, stream>>>(src, dst, ev, denom, h1, agg1, E);

    // h1 <- relu(agg1 + b1)
    bias_act<<<cdiv(N * H, B), B, 0, stream>>>(agg1, b1, h1, N * H, H, /*relu=*/1);

    // ---------------- layer 2 ----------------
    gemm_wmma_strip<2><<<cdiv(N, 128), B, H * O * sizeof(float), stream>>>(h1, W2, h2, N, H);

    node_scores<<<cdiv(N * 32, B), B, 0, stream>>>(h2, a_src2, a_dst2, ls, ld, N, O);

    fill_f32<<<cdiv(N, B), B, 0, stream>>>(lmax, -INFINITY, N);
    fill_f32<<<cdiv(N, B), B, 0, stream>>>(denom, 0.0f, N);
    fill_f32<<<cdiv(N * O, B), B, 0, stream>>>(out, 0.0f, N * O);

    edge_logit_max<<<cdiv(E, B), B, 0, stream>>>(src, dst, ls, ld, ev, lmax, E);
    edge_expsum<<<cdiv(E, B), B, 0, stream>>>(dst, lmax, ev, denom, E);
    edge_aggregate<32><<<cdiv(E * (O / 4), B), B, 0, stream>>>(src, dst, ev, denom, h2, out, E);

    // out <- out + b2 (in-place, no relu)
    bias_act<<<cdiv(N * O, B), B, 0, stream>>>(out, b2, out, N * O, O, /*relu=*/0);
}